// SUB_Independent_Loss_44796508897297
// MI455X (gfx1250) — compile-verified
//
#include <hip/hip_runtime.h>

typedef __attribute__((ext_vector_type(2))) float v2f;
typedef __attribute__((ext_vector_type(8))) float v8f;

#define NROWS 256
#define DDIM  1024

// ---------------------------------------------------------------------------
// Phase 1: sim12 = f1*f2^T, sim11 = f1*f1^T, sim22 = f2*f2^T via
// V_WMMA_F32_16X16X4_F32 (exact fp32 matrix pipe). One wave per 16x16 tile.
//
// A-frag layout (16x4 f32, MxK): lane l, vgpr v holds A[l%16][2*(l/16)+v]
// B-frag layout (4x16 f32, KxN): lane l, vgpr v holds B[2*(l/16)+v][l%16]
// For C = X * Y^T both frags load identically from row-major X and Y:
//   lane l reads a contiguous float2 at row (l%16), cols 2*(l/16)..+1.
// C layout: vgpr j, lane l -> C[M = j + 8*(l/16)][N = l%16]
// ---------------------------------------------------------------------------
__global__ __launch_bounds__(32) void gemm_tiles_f32wmma(
    const float* __restrict__ f1, const float* __restrict__ f2,
    float* __restrict__ sims) {
  const int mt = blockIdx.x, nt = blockIdx.y, z = blockIdx.z;
  const float* X;
  const float* Y;
  float* out;
  if (z == 0)      { X = f1; Y = f2; out = sims; }
  else if (z == 1) { X = f1; Y = f1; out = sims + NROWS * NROWS; }
  else             { X = f2; Y = f2; out = sims + 2 * NROWS * NROWS; }

  const int lane = threadIdx.x;        // 0..31
  const int r = lane & 15;             // row within 16-row tile
  const int h = lane >> 4;             // half-wave selector (K pairs / M+8)

  const float* arow = X + (size_t)(mt * 16 + r) * DDIM + 2 * h;
  const float* brow = Y + (size_t)(nt * 16 + r) * DDIM + 2 * h;

  v8f c0 = {};
  v8f c1 = {};
  for (int k = 0; k < DDIM; k += 8) {
    v2f a0 = *(const v2f*)(arow + k);
    v2f b0 = *(const v2f*)(brow + k);
    v2f a1 = *(const v2f*)(arow + k + 4);
    v2f b1 = *(const v2f*)(brow + k + 4);
    // 8 args: (neg_a, A, neg_b, B, c_mod, C, reuse_a, reuse_b)
    c0 = __builtin_amdgcn_wmma_f32_16x16x4_f32(false, a0, false, b0,
                                               (short)0, c0, false, false);
    c1 = __builtin_amdgcn_wmma_f32_16x16x4_f32(false, a1, false, b1,
                                               (short)0, c1, false, false);
  }

#pragma unroll
  for (int j = 0; j < 8; ++j) {
    const int row = mt * 16 + j + 8 * h;
    out[(size_t)row * NROWS + nt * 16 + r] = c0[j] + c1[j];
  }
}

// ---------------------------------------------------------------------------
// Phase 2: pairwise ranking loss. One 256-thread block per (config, row).
//   cfg 0: sim12 rows    / mask_cross rows    / sel = mask_sents1
//   cfg 1: sim12 columns / mask_cross columns / sel = mask_sents2
//   cfg 2: sim11 rows    / mask1 rows         / sel = mask_sents1
//   cfg 3: sim22 rows    / mask2 rows         / sel = mask_sents2
// loss_i = sel_i * sum_{j:pos} sum_{k:neg} softplus(s_k - s_j)
// pairs_i = sel_i * npos_i * nneg_i
// ---------------------------------------------------------------------------
__global__ __launch_bounds__(256) void pairwise_loss(
    const float* __restrict__ sims,
    const int* __restrict__ mc, const int* __restrict__ m1,
    const int* __restrict__ m2,
    const unsigned char* __restrict__ s1, const unsigned char* __restrict__ s2,
    float* __restrict__ loss_acc, unsigned long long* __restrict__ pair_acc) {
  const int cfg = blockIdx.x >> 8;
  const int i = blockIdx.x & 255;
  const int t = threadIdx.x;

  __shared__ float sh_s[NROWS];
  __shared__ float sh_mneg[NROWS];
  __shared__ int   sh_mpos[NROWS];
  __shared__ float sh_red[NROWS];

  const float* sim12 = sims;
  const float* sim11 = sims + NROWS * NROWS;
  const float* sim22 = sims + 2 * NROWS * NROWS;

  int sel, mval;
  float sv;
  switch (cfg) {
    case 0:  sel = s1[i]; sv = sim12[i * NROWS + t]; mval = mc[i * NROWS + t]; break;
    case 1:  sel = s2[i]; sv = sim12[t * NROWS + i]; mval = mc[t * NROWS + i]; break;
    case 2:  sel = s1[i]; sv = sim11[i * NROWS + t]; mval = m1[i * NROWS + t]; break;
    default: sel = s2[i]; sv = sim22[i * NROWS + t]; mval = m2[i * NROWS + t]; break;
  }
  if (!sel) return;  // uniform per block: contributes 0 loss, 0 pairs

  sh_s[t] = sv;
  sh_mneg[t] = mval ? 0.0f : 1.0f;
  sh_mpos[t] = mval ? 1 : 0;
  __syncthreads();

  const float st = sh_s[t];
  float acc = 0.0f;
  for (int k = 0; k < NROWS; ++k) {
    const float x = sh_s[k] - st;
    // stable softplus: max(x,0) + log(1 + exp(-|x|)); exp arg <= 0 so 1+e in [1,2]
    const float sp = fmaxf(x, 0.0f) + __logf(1.0f + __expf(-fabsf(x)));
    acc = fmaf(sh_mneg[k], sp, acc);
  }
  acc *= (float)sh_mpos[t];  // only positive rows j contribute

  sh_red[t] = acc;
  __syncthreads();
  for (int off = 128; off > 0; off >>= 1) {
    if (t < off) sh_red[t] += sh_red[t + off];
    __syncthreads();
  }

  if (t == 0) {
    int pos = 0;
    for (int k = 0; k < NROWS; ++k) pos += sh_mpos[k];
    const unsigned long long pairs =
        (unsigned long long)pos * (unsigned long long)(NROWS - pos);
    if (pairs) atomicAdd(pair_acc, pairs);
    atomicAdd(loss_acc, sh_red[0]);
  }
}

__global__ void init_acc(float* loss_acc, unsigned long long* pair_acc) {
  *loss_acc = 0.0f;
  *pair_acc = 0ull;
}

__global__ void finalize(const float* __restrict__ loss_acc,
                         const unsigned long long* __restrict__ pair_acc,
                         float* __restrict__ out) {
  const unsigned long long p = *pair_acc;
  const float l = *loss_acc;
  out[0] = (p > 0ull) ? (l / (float)p) : l;
}

extern "C" void kernel_launch(void* const* d_in, const int* in_sizes, int n_in,
                              void* d_out, int out_size, void* d_ws, size_t ws_size,
                              hipStream_t stream) {
  const float* f1 = (const float*)d_in[0];          // [256,1024] f32
  const float* f2 = (const float*)d_in[1];          // [256,1024] f32
  const int* mc = (const int*)d_in[2];              // [256,256] i32
  const int* m1 = (const int*)d_in[3];              // [256,256] i32
  const int* m2 = (const int*)d_in[4];              // [256,256] i32
  const unsigned char* s1 = (const unsigned char*)d_in[5];  // [256] bool
  const unsigned char* s2 = (const unsigned char*)d_in[6];  // [256] bool

  float* sims = (float*)d_ws;  // 3 * 256*256 f32 = 768 KB
  float* loss_acc = (float*)((char*)d_ws + 3ull * NROWS * NROWS * sizeof(float));
  unsigned long long* pair_acc = (unsigned long long*)(loss_acc + 2);  // 8B aligned

  dim3 ggrid(16, 16, 3);
  gemm_tiles_f32wmma<<<ggrid, 32, 0, stream>>>(f1, f2, sims);
  init_acc<<<1, 1, 0, stream>>>(loss_acc, pair_acc);
  pairwise_loss<<<4 * NROWS, NROWS, 0, stream>>>(sims, mc, m1, m2, s1, s2,
                                                 loss_acc, pair_acc);
  finalize<<<1, 1, 0, stream>>>(loss_acc, pair_acc, (float*)d_out);
}